// AbstractFullyConnected_58531814309987
// MI455X (gfx1250) — compile-verified
//
#include <hip/hip_runtime.h>

typedef __attribute__((ext_vector_type(2))) float v2f;
typedef __attribute__((ext_vector_type(8))) float v8f;

#define NORM_MEAN 0.1307f
#define NORM_STD  0.3081f
#define VEC_STRIDE 4096       // element stride between x / low / high planes in global scratch
#define WAVES 8               // waves per block (wave32 -> 256 threads)
#define NSPLIT 4              // K-dimension split factor (occupancy / latency hiding)
#define CHUNK_MAX 1024        // max K-chunk per block (4096 / NSPLIT)

// Normalize x, low, high into the first ping-pong buffer (x plane, low plane, high plane).
__global__ void norm_kernel(const float* __restrict__ x,
                            const float* __restrict__ low,
                            const float* __restrict__ high,
                            float* __restrict__ buf, int n) {
    int i = blockIdx.x * blockDim.x + threadIdx.x;
    if (i < n) {
        buf[i]                = (x[i]    - NORM_MEAN) / NORM_STD;
        buf[VEC_STRIDE + i]   = (low[i]  - NORM_MEAN) / NORM_STD;
        buf[2*VEC_STRIDE + i] = (high[i] - NORM_MEAN) / NORM_STD;
    }
}

// Partial abstract-linear pass over one K-chunk. Each wave owns one 16-output tile;
// blockIdx.y selects the K split. Accumulation via V_WMMA_F32_16X16X4_F32:
//   D += A_pos x max(W,0) ; D += A_neg x min(W,0)
// A rows: 0=x, 1=low, 2=high (A_neg swaps rows 1/2), rows 3..15 zero.
__launch_bounds__(32 * WAVES)
__global__ void layer_partial_kernel(const float* __restrict__ W,
                                     const float* __restrict__ in_vecs, // x/low/high, stride VEC_STRIDE
                                     float* __restrict__ partial,       // [NSPLIT][3][4096]
                                     int in_dim, int out_dim, int chunk) {
    __shared__ float lds[3 * CHUNK_MAX + 16];

    const int split = blockIdx.y;
    const int kbeg  = split * chunk;

    // Stage this block's K-chunk of the three input vectors into LDS.
    for (int i = threadIdx.x; i < chunk; i += blockDim.x) {
        lds[i]             = in_vecs[kbeg + i];
        lds[chunk + i]     = in_vecs[VEC_STRIDE + kbeg + i];
        lds[2 * chunk + i] = in_vecs[2 * VEC_STRIDE + kbeg + i];
    }
    if (threadIdx.x < 16) lds[3 * chunk + threadIdx.x] = 0.0f; // zero pad for A rows 3..15
    __syncthreads();

    const int lane   = threadIdx.x & 31;
    const int wave   = threadIdx.x >> 5;
    const int ntiles = (out_dim + 15) >> 4;
    const int tile   = blockIdx.x * WAVES + wave;
    const bool valid_tile = tile < ntiles;
    const int nb = (valid_tile ? tile : 0) * 16;

    const int m  = lane & 15;        // A row / B column within tile
    const int kq = (lane >> 4) * 2;  // K sub-offset per ISA fragment layout

    // Per-lane LDS bases for pos / neg A fragments. Lanes m>=3 read the zero pad
    // with zero stride so the K loop stays branch-free and EXEC stays all-ones.
    const float* ap;
    const float* an;
    int astep;
    if (m == 0)      { ap = lds;             an = lds;             astep = 4; }
    else if (m == 1) { ap = lds + chunk;     an = lds + 2*chunk;   astep = 4; }
    else if (m == 2) { ap = lds + 2*chunk;   an = lds + chunk;     astep = 4; }
    else             { ap = lds + 3*chunk;   an = lds + 3*chunk;   astep = 0; }
    ap += kq;
    an += kq;

    // B[k][n] = W[nb+n][k] -> lane loads W[nb+m][kbeg+kk+kq .. +1].
    int wrow = nb + m;
    if (wrow >= out_dim) wrow = out_dim - 1;  // clamp OOB rows (their D columns never stored)
    const float* wp = W + (size_t)wrow * (size_t)in_dim + kbeg + kq;

    v8f acc = {};
#pragma unroll 4
    for (int kk = 0; kk < chunk; kk += 4) {
        v2f a_p = *(const v2f*)ap;
        v2f a_n = *(const v2f*)an;
        v2f w   = *(const v2f*)wp;
        v2f w_pos, w_neg;
        w_pos.x = fmaxf(w.x, 0.0f); w_pos.y = fmaxf(w.y, 0.0f);
        w_neg.x = fminf(w.x, 0.0f); w_neg.y = fminf(w.y, 0.0f);
        acc = __builtin_amdgcn_wmma_f32_16x16x4_f32(false, a_p, false, w_pos,
                                                    (short)0, acc, false, false);
        acc = __builtin_amdgcn_wmma_f32_16x16x4_f32(false, a_n, false, w_neg,
                                                    (short)0, acc, false, false);
        ap += astep;
        an += astep;
        wp += 4;
    }

    // Lanes 0..15 hold D rows 0..2 in acc[0..2] (row p = acc index, N = lane).
    if (valid_tile && lane < 16) {
        int n = nb + lane;
        if (n < out_dim) {
            float* p = partial + (size_t)split * 3 * VEC_STRIDE;
            p[n]                  = acc[0];  // concrete pass
            p[VEC_STRIDE + n]     = acc[1];  // lower bound
            p[2 * VEC_STRIDE + n] = acc[2];  // upper bound
        }
    }
}

// Sum the NSPLIT K-partials (fixed order -> deterministic), add bias, apply the
// DeepPoly ReLU relaxation (unless last layer), and emit the next x/low/high planes.
__global__ void reduce_transform_kernel(const float* __restrict__ partial,
                                        const float* __restrict__ bias,
                                        float* __restrict__ out_x,
                                        float* __restrict__ out_low,
                                        float* __restrict__ out_high,
                                        int out_dim, int last) {
    int n = blockIdx.x * blockDim.x + threadIdx.x;
    if (n >= out_dim) return;
    float xr = 0.0f, lo = 0.0f, hi = 0.0f;
#pragma unroll
    for (int s = 0; s < NSPLIT; ++s) {
        const float* p = partial + (size_t)s * 3 * VEC_STRIDE;
        xr += p[n];
        lo += p[VEC_STRIDE + n];
        hi += p[2 * VEC_STRIDE + n];
    }
    float b = bias[n];
    xr += b; lo += b; hi += b;
    if (!last) {
        float denom    = hi - lo;
        float ub_slope = hi / (denom + 1e-6f);
        float ub_int   = lo * hi / denom;
        hi = ub_slope * hi + ub_int;
        lo = 0.0f;
        xr = fmaxf(xr, 0.0f);
    }
    out_x[n]    = xr;
    out_low[n]  = lo;
    out_high[n] = hi;
}

extern "C" void kernel_launch(void* const* d_in, const int* in_sizes, int n_in,
                              void* d_out, int out_size, void* d_ws, size_t ws_size,
                              hipStream_t stream) {
    (void)in_sizes; (void)n_in; (void)out_size; (void)ws_size;
    const float* x    = (const float*)d_in[0];
    const float* low  = (const float*)d_in[1];
    const float* high = (const float*)d_in[2];
    const float* W[4] = {(const float*)d_in[3], (const float*)d_in[4],
                         (const float*)d_in[5], (const float*)d_in[6]};
    const float* b[4] = {(const float*)d_in[7], (const float*)d_in[8],
                         (const float*)d_in[9], (const float*)d_in[10]};

    float* bufA    = (float*)d_ws;               // 3 * 4096 floats
    float* bufB    = bufA + 3 * VEC_STRIDE;      // 3 * 4096 floats
    float* partial = bufB + 3 * VEC_STRIDE;      // NSPLIT * 3 * 4096 floats
    const int sizes[5] = {784, 4096, 4096, 4096, 10};

    norm_kernel<<<(784 + 255) / 256, 256, 0, stream>>>(x, low, high, bufA, 784);

    float* cur = bufA;
    float* nxt = bufB;
    for (int i = 0; i < 4; ++i) {
        const int in_dim  = sizes[i];
        const int out_dim = sizes[i + 1];
        const int last    = (i == 3);
        const int chunk   = in_dim / NSPLIT;       // 196 or 1024, both multiples of 4
        const int ntiles  = (out_dim + 15) / 16;
        const int blocks  = (ntiles + WAVES - 1) / WAVES;

        dim3 grid(blocks, NSPLIT);
        layer_partial_kernel<<<grid, 32 * WAVES, 0, stream>>>(
            W[i], cur, partial, in_dim, out_dim, chunk);

        float* ox, *ol, *oh;
        if (!last) {
            ox = nxt; ol = nxt + VEC_STRIDE; oh = nxt + 2 * VEC_STRIDE;
        } else {
            float* out = (float*)d_out;
            ox = out; ol = out + 10; oh = out + 20;
        }
        reduce_transform_kernel<<<(out_dim + 255) / 256, 256, 0, stream>>>(
            partial, b[i], ox, ol, oh, out_dim, last);

        if (!last) { float* t = cur; cur = nxt; nxt = t; }
    }
}